// KNNClassifier_70643622084869
// MI455X (gfx1250) — compile-verified
//
#include <hip/hip_runtime.h>

typedef __attribute__((ext_vector_type(2))) float v2f;
typedef __attribute__((ext_vector_type(4))) float v4f;
typedef __attribute__((ext_vector_type(8))) float v8f;

#if defined(__has_builtin)
#  if __has_builtin(__builtin_amdgcn_global_load_async_to_lds_b32)
#    define HAS_ASYNC_LDS 1
#  endif
#endif
#ifndef HAS_ASYNC_LDS
#  define HAS_ASYNC_LDS 0
#endif

#define GPTR(p) ((__attribute__((address_space(1))) int*)(void*)(p))
#define LPTR(p) ((__attribute__((address_space(3))) int*)(void*)(p))

// Map float -> uint32 so unsigned compare == float compare (total order).
__device__ __forceinline__ unsigned fkey(float f) {
    unsigned u = __float_as_uint(f);
    return (u & 0x80000000u) ? ~u : (u | 0x80000000u);
}

// ---------------------------------------------------------------------------
// Kernel 1: fp32 WMMA GEMM  sim[M,N] = A[M,D] * B[D,N]
// Block = 256 threads (8 wave32), block tile 64x128, wave tile 16x64.
// B tiles (16 K-rows x 128 cols) staged through a double-buffered LDS region,
// column-major with stride 18 so each B fragment is ONE 8B-aligned,
// conflict-free ds_load_b64. Inner loop has no bounds checks (tail columns
// are clamped during staging; their results are simply never stored).
// Staging uses gfx1250 async global->LDS copies (ASYNCcnt) when available.
// ---------------------------------------------------------------------------
#define LDS_STRIDE 18   // words per column: 16 K values + 2 pad (keeps b64 aligned, banks spread)

__global__ __launch_bounds__(256) void knn_gemm_wmma(
    const float* __restrict__ A, const float* __restrict__ Bm,
    float* __restrict__ sim, int D, int N, int Mrows)
{
    __shared__ float ldsB[2][128 * LDS_STRIDE];

    const int tid   = threadIdx.x;
    const int lane  = tid & 31;
    const int wave  = tid >> 5;
    const int laneM = lane & 15;
    const int kh    = lane >> 4;            // 0: K{0,1} half, 1: K{2,3} half

    const int rowBase  = blockIdx.x * 64 + (wave >> 1) * 16;
    const int colBase  = blockIdx.y * 128 + (wave & 1) * 64;
    const int blockCol = blockIdx.y * 128;

    int arow = rowBase + laneM;
    if (arow >= Mrows) arow = Mrows - 1;    // clamp (stores are guarded)
    const float* __restrict__ Ap = A + (size_t)arow * D;

    // staging map: this thread owns column sc (fixed) and rows sr0 + {0,2,..,14}
    const int sc  = tid & 127;
    const int sr0 = tid >> 7;               // 0 or 1
    int gcol = blockCol + sc;
    if (gcol >= N) gcol = N - 1;            // clamp tail (values unused)
    const float* __restrict__ Bcol = Bm + gcol;

    v8f acc0 = {}, acc1 = {}, acc2 = {}, acc3 = {};

    const int KT = D >> 4;                  // number of 16-deep K tiles
#if !HAS_ASYNC_LDS
    float stg[8];
#endif

    // ---- prologue: stage tile 0 into buffer 0 ----
#if HAS_ASYNC_LDS
    #pragma unroll
    for (int i = 0; i < 8; ++i) {
        const int r = sr0 + 2 * i;
        __builtin_amdgcn_global_load_async_to_lds_b32(
            GPTR(Bcol + (size_t)r * N),
            LPTR(&ldsB[0][sc * LDS_STRIDE + r]),
            0, 0);
    }
    asm volatile("s_wait_asynccnt 0x0" ::: "memory");
#else
    #pragma unroll
    for (int i = 0; i < 8; ++i) {
        const int r = sr0 + 2 * i;
        ldsB[0][sc * LDS_STRIDE + r] = Bcol[(size_t)r * N];
    }
#endif
    __syncthreads();

    int cur = 0;
    for (int kt = 0; kt < KT; ++kt) {
        const int nxt = cur ^ 1;
        const int kb  = kt << 4;

        // ---- issue staging of next tile (overlapped with compute) ----
        if (kt + 1 < KT) {
#if HAS_ASYNC_LDS
            #pragma unroll
            for (int i = 0; i < 8; ++i) {
                const int r = sr0 + 2 * i;
                __builtin_amdgcn_global_load_async_to_lds_b32(
                    GPTR(Bcol + (size_t)(kb + 16 + r) * N),
                    LPTR(&ldsB[nxt][sc * LDS_STRIDE + r]),
                    0, 0);
            }
#else
            #pragma unroll
            for (int i = 0; i < 8; ++i) {
                const int r = sr0 + 2 * i;
                stg[i] = Bcol[(size_t)(kb + 16 + r) * N];
            }
#endif
        }

        // ---- compute: 16 WMMAs on current tile, no guards ----
        const float* __restrict__ lb = &ldsB[cur][((wave & 1) * 64 + laneM) * LDS_STRIDE + 2 * kh];
        #pragma unroll
        for (int kk = 0; kk < 16; kk += 4) {
            const v2f a = *(const v2f*)(Ap + kb + kk + 2 * kh);
            const v2f b0 = *(const v2f*)(lb + 0 * 16 * LDS_STRIDE + kk);
            const v2f b1 = *(const v2f*)(lb + 1 * 16 * LDS_STRIDE + kk);
            const v2f b2 = *(const v2f*)(lb + 2 * 16 * LDS_STRIDE + kk);
            const v2f b3 = *(const v2f*)(lb + 3 * 16 * LDS_STRIDE + kk);
            acc0 = __builtin_amdgcn_wmma_f32_16x16x4_f32(false, a, false, b0, (short)0, acc0, false, false);
            acc1 = __builtin_amdgcn_wmma_f32_16x16x4_f32(false, a, false, b1, (short)0, acc1, false, false);
            acc2 = __builtin_amdgcn_wmma_f32_16x16x4_f32(false, a, false, b2, (short)0, acc2, false, false);
            acc3 = __builtin_amdgcn_wmma_f32_16x16x4_f32(false, a, false, b3, (short)0, acc3, false, false);
        }

        // ---- make next buffer visible ----
        if (kt + 1 < KT) {
#if HAS_ASYNC_LDS
            asm volatile("s_wait_asynccnt 0x0" ::: "memory");
            __syncthreads();
#else
            __syncthreads();
            #pragma unroll
            for (int i = 0; i < 8; ++i) {
                const int r = sr0 + 2 * i;
                ldsB[nxt][sc * LDS_STRIDE + r] = stg[i];
            }
            __syncthreads();
#endif
            cur = nxt;
        }
    }

    // ---- epilogue: guarded stores ----
    #pragma unroll
    for (int t = 0; t < 4; ++t) {
        const int col = colBase + t * 16 + laneM;
        if (col >= N) continue;
        const v8f c = (t == 0) ? acc0 : (t == 1) ? acc1 : (t == 2) ? acc2 : acc3;
        #pragma unroll
        for (int i = 0; i < 8; ++i) {
            const int row = rowBase + i + 8 * kh;
            if (row < Mrows) sim[(size_t)row * N + col] = c[i];
        }
    }
}

// ---------------------------------------------------------------------------
// Kernel 2: per-row exact top-k threshold via 3-pass radix select
// (11/11/10 bits, 2048-bin LDS histogram), then exp-weighted class vote into
// LDS (ds_add_f32), tie-capped to exactly k. One 256-thread block per row.
// ---------------------------------------------------------------------------
__global__ __launch_bounds__(256) void knn_select_vote(
    const float* __restrict__ sim, const int* __restrict__ labels,
    float* __restrict__ scores, int N,
    const int* __restrict__ kPtr, const int* __restrict__ cPtr, int rowOff)
{
    __shared__ unsigned hist[2048];
    __shared__ float    cls[1024];
    __shared__ unsigned s_prefix, s_need, s_ties;

    const int tid = threadIdx.x;
    const float* __restrict__ rowp = sim + (size_t)blockIdx.x * N;

    if (tid == 0) { s_prefix = 0u; s_need = (unsigned)(*kPtr); }
    __syncthreads();

    const int SH[3]   = {21, 10, 0};
    const int BITS[3] = {11, 11, 10};

    for (int p = 0; p < 3; ++p) {
        const int shift = SH[p];
        const unsigned nb = 1u << BITS[p];
        for (unsigned i = tid; i < nb; i += 256) hist[i] = 0u;
        __syncthreads();
        const int hiBit = shift + BITS[p];
        const unsigned hiMask = (hiBit >= 32) ? 0u : (0xFFFFFFFFu << hiBit);
        const unsigned pfx = s_prefix;
        for (int c = tid; c < N; c += 256) {
            const unsigned key = fkey(rowp[c]);
            if ((key & hiMask) == pfx)
                atomicAdd(&hist[(key >> shift) & (nb - 1u)], 1u);
        }
        __syncthreads();
        if (tid == 0) {
            unsigned need = s_need, cum = 0u;
            for (int b = (int)nb - 1; b >= 0; --b) {
                const unsigned h = hist[b];
                if (cum + h >= need) {
                    s_prefix = pfx | ((unsigned)b << shift);
                    s_need   = need - cum;
                    break;
                }
                cum += h;
            }
        }
        __syncthreads();
    }

    // vote pass
    for (int c = tid; c < 1024; c += 256) cls[c] = 0.0f;
    if (tid == 0) s_ties = 0u;
    __syncthreads();

    const unsigned T = s_prefix;
    const unsigned tiesNeed = s_need;
    for (int c = tid; c < N; c += 256) {
        const float v = rowp[c];
        const unsigned key = fkey(v);
        if (key >= T) {
            bool take = (key > T);
            if (!take) {                       // exactly at threshold: cap at k
                const unsigned p = atomicAdd(&s_ties, 1u);
                take = (p < tiesNeed);
            }
            if (take) atomicAdd(&cls[labels[c]], __expf(v * 10.0f)); // 1/KNN_T
        }
    }
    __syncthreads();

    const int C = *cPtr;
    const size_t grow = (size_t)(rowOff + (int)blockIdx.x);
    for (int c = tid; c < C; c += 256) scores[grow * C + c] = cls[c];
}

// ---------------------------------------------------------------------------
// Kernel 3: per-row descending stable argsort of C class scores.
// 1024-wide LDS bitonic sort on key = (~sortable(score) << 32) | class.
// ---------------------------------------------------------------------------
__global__ __launch_bounds__(256) void knn_argsort(
    const float* __restrict__ scores, float* __restrict__ out,
    const int* __restrict__ cPtr)
{
    __shared__ unsigned long long keys[1024];
    const int C = *cPtr;
    const int row = blockIdx.x;

    for (int i = threadIdx.x; i < 1024; i += 256) {
        if (i < C) {
            const unsigned us = fkey(scores[(size_t)row * C + i]);
            keys[i] = ((unsigned long long)(~us) << 32) | (unsigned)i;
        } else {
            keys[i] = 0xFFFFFFFF00000000ull | (unsigned)i;   // pad sorts last
        }
    }
    __syncthreads();

    for (unsigned size = 2; size <= 1024; size <<= 1) {
        for (unsigned stride = size >> 1; stride > 0; stride >>= 1) {
            for (unsigned j = threadIdx.x; j < 1024; j += 256) {
                const unsigned ixj = j ^ stride;
                if (ixj > j) {
                    const bool asc = ((j & size) == 0);
                    const unsigned long long a = keys[j], b = keys[ixj];
                    if ((a > b) == asc) { keys[j] = b; keys[ixj] = a; }
                }
            }
            __syncthreads();
        }
    }

    for (int i = threadIdx.x; i < C; i += 256)
        out[(size_t)row * C + i] = (float)(unsigned)(keys[i] & 0xFFFFFFFFu);
}

// ---------------------------------------------------------------------------
extern "C" void kernel_launch(void* const* d_in, const int* in_sizes, int n_in,
                              void* d_out, int out_size, void* d_ws, size_t ws_size,
                              hipStream_t stream) {
    const float* feature = (const float*)d_in[0];
    const float* bank    = (const float*)d_in[1];
    const int*   labels  = (const int*)d_in[2];
    const int*   cPtr    = (const int*)d_in[3];   // num_classes
    const int*   kPtr    = (const int*)d_in[4];   // knn_k

    const int N = in_sizes[2];                    // 100000
    const int D = in_sizes[1] / N;                // 1024
    const int B = in_sizes[0] / D;                // 1024

    // workspace layout: [class scores: B*1024 floats][sim slab: slab*N floats]
    float* scores = (float*)d_ws;
    const size_t scoresBytes = (size_t)B * 1024 * sizeof(float);
    float* sim = (float*)((char*)d_ws + scoresBytes);

    const size_t avail = (ws_size > scoresBytes) ? (ws_size - scoresBytes) : 0;
    long long maxRows = (long long)(avail / ((size_t)N * sizeof(float)));
    int slab = (int)((maxRows / 64) * 64);
    if (slab < 64) slab = 64;
    if (slab > B)  slab = B;

    const int colChunks = (N + 127) / 128;

    for (int r0 = 0; r0 < B; r0 += slab) {
        const int rows = (B - r0 < slab) ? (B - r0) : slab;
        dim3 grid((rows + 63) / 64, colChunks);
        knn_gemm_wmma<<<grid, 256, 0, stream>>>(feature + (size_t)r0 * D, bank,
                                                sim, D, N, rows);
        knn_select_vote<<<rows, 256, 0, stream>>>(sim, labels, scores, N,
                                                  kPtr, cPtr, r0);
    }
    knn_argsort<<<B, 256, 0, stream>>>(scores, (float*)d_out, cPtr);
}